// GraphConv_55018531062594
// MI455X (gfx1250) — compile-verified
//
#include <hip/hip_runtime.h>
#include <math.h>
#include <limits.h>

#define CDIM 64
#define NFACT 4
#define NRELM1 9
#define TEMPI 0.2f

typedef __attribute__((ext_vector_type(2))) float v2f;
typedef __attribute__((ext_vector_type(8))) float v8f;

// async-to-LDS builtin operand types (from clang diagnostic):
//   (v4i addrspace(1)*, v4i addrspace(3)*, imm int offset, imm int cpol)
typedef int v4i __attribute__((__vector_size__(16)));
typedef __attribute__((address_space(1))) v4i* as1_v4i;
typedef __attribute__((address_space(3))) v4i* as3_v4i;

#if __has_builtin(__builtin_amdgcn_global_load_async_to_lds_b128)
#define USE_ASYNC_LDS 1
#else
#define USE_ASYNC_LDS 0
#endif

__device__ __forceinline__ void wait_async0() {
#if __has_builtin(__builtin_amdgcn_s_wait_asynccnt)
  __builtin_amdgcn_s_wait_asynccnt(0);
#else
  asm volatile("s_wait_asynccnt 0x0" ::: "memory");
#endif
}

__device__ __forceinline__ int f2ord(float f) {
  int i = __float_as_int(f);
  return (i >= 0) ? i : (i ^ 0x7FFFFFFF);
}
__device__ __forceinline__ float ord2f(int i) {
  return __int_as_float((i >= 0) ? i : (i ^ 0x7FFFFFFF));
}

// ---------------------------------------------------------------------------
// Tiny precompute. All intermediates (rel_proj, Bmat) live in LDS.
// Outputs: Bswz[16*32 v2f] (WMMA-B-operand-ordered), relnorm2[16],
//          disen_w[4x64], scalar cor.
// Bmat[a,j] = 1/(2*sqrt(32)) * sum_c kgW[a,c] * rel_proj[j,c]; j>=9 -> 0
// Bswz[k4*32+lane] = { Bmat[4*k4+2*khalf, col], Bmat[4*k4+2*khalf+1, col] }
// ---------------------------------------------------------------------------
__global__ void precompute_small(const float* __restrict__ weight,
                                 const float* __restrict__ kgW,
                                 const float* __restrict__ dwa,
                                 float* __restrict__ Bswz,
                                 float* __restrict__ relnorm2,
                                 float* __restrict__ disenw,
                                 float* __restrict__ cor_out) {
  __shared__ float srel[NRELM1 * CDIM];   // rel_proj = weight @ kgW
  __shared__ float sB[CDIM * 16];         // Bmat
  const int t = threadIdx.x;              // blockDim.x == 256
  for (int idx = t; idx < NRELM1 * CDIM; idx += 256) {
    int r = idx / CDIM, c = idx % CDIM;
    float s = 0.f;
    for (int b = 0; b < CDIM; ++b) s += weight[r * CDIM + b] * kgW[b * CDIM + c];
    srel[idx] = s;
  }
  __syncthreads();
  const float invsc = 1.0f / (2.0f * sqrtf((float)(CDIM / 2)));
  for (int idx = t; idx < CDIM * 16; idx += 256) {
    int a = idx / 16, j = idx % 16;
    float s = 0.f;
    if (j < NRELM1) {
      for (int c = 0; c < CDIM; ++c) s += kgW[a * CDIM + c] * srel[j * CDIM + c];
      s *= invsc;
    }
    sB[idx] = s;
  }
  __syncthreads();
  for (int idx = t; idx < 512; idx += 256) {  // 16 k4-steps x 32 lanes
    int k4 = idx >> 5, lane = idx & 31;
    int col = lane & 15, khalf = lane >> 4;
    int kk = 4 * k4 + 2 * khalf;
    Bswz[2 * idx]     = sB[kk * 16 + col];
    Bswz[2 * idx + 1] = sB[(kk + 1) * 16 + col];
  }
  if (t < 16) {
    float s = 0.f;
    if (t < NRELM1)
      for (int c = 0; c < CDIM; ++c) { float w = weight[t * CDIM + c]; s += w * w; }
    relnorm2[t] = s;
  }
  for (int idx = t; idx < NFACT * CDIM; idx += 256) {
    int f = idx / CDIM, c = idx % CDIM;
    float m = -1e30f;
    for (int r = 0; r < NRELM1; ++r) m = fmaxf(m, dwa[f * NRELM1 + r]);
    float den = 0.f;
    for (int r = 0; r < NRELM1; ++r) den += expf(dwa[f * NRELM1 + r] - m);
    float s = 0.f;
    for (int r = 0; r < NRELM1; ++r)
      s += (expf(dwa[f * NRELM1 + r] - m) / den) * weight[r * CDIM + c];
    disenw[idx] = s;
  }
  if (t == 0) {
    float rowsum[NFACT];
    for (int f = 0; f < NFACT; ++f) {
      float s = 0.f;
      for (int j = 0; j < NRELM1; ++j) s += dwa[f * NRELM1 + j];
      rowsum[f] = s;
    }
    float cor = 0.f;
    for (int i = 0; i < NRELM1; ++i) {
      float nrm2 = 0.f, ttl = 0.f;
      for (int f = 0; f < NFACT; ++f) {
        float v = dwa[f * NRELM1 + i];
        nrm2 += v * v;
        ttl += v * rowsum[f];
      }
      float nrm = sqrtf(nrm2);
      float pos = 0.f;
      for (int f = 0; f < NFACT; ++f) {
        float v = dwa[f * NRELM1 + i] / nrm;
        pos += v * v;
      }
      cor += (ttl - pos) / TEMPI;
    }
    *cor_out = cor;
  }
}

// ---------------------------------------------------------------------------
// WMMA f32 16x16x4: M[n_ent,16] = entity_emb[n_ent,64] @ B[64,16]
// One wave per 16-row tile. The 16x64 A tile is a contiguous 4KB block:
// stage into LDS (async-to-LDS when available), then a branch-free K loop of
// 16 v_wmma_f32_16x16x4_f32 reading v2f A from LDS and pre-swizzled B (b64).
// ---------------------------------------------------------------------------
__global__ void __launch_bounds__(32)
wmma_scores(const float* __restrict__ E, const float* __restrict__ Bswz,
            float* __restrict__ M, int n_ent) {
  __shared__ float atile[16 * CDIM];
  const int lane = threadIdx.x & 31;
  const int r0 = blockIdx.x * 16;
  const int col = lane & 15;
  const int khalf = lane >> 4;

  if (r0 + 16 <= n_ent) {
#if USE_ASYNC_LDS
    const char* src = (const char*)(E + (size_t)r0 * CDIM);
#pragma unroll
    for (int j = 0; j < 8; ++j) {
      const int byteoff = (j * 32 + lane) * 16;
      __builtin_amdgcn_global_load_async_to_lds_b128(
          (as1_v4i)(src + byteoff),
          (as3_v4i)((char*)atile + byteoff),
          0, 0);
    }
    wait_async0();
#else
    const float4* src = (const float4*)(E + (size_t)r0 * CDIM);
    float4* dst = (float4*)atile;
#pragma unroll
    for (int j = 0; j < 8; ++j) dst[j * 32 + lane] = src[j * 32 + lane];
#endif
  } else {
    const float4* src = (const float4*)(E + (size_t)r0 * CDIM);
    float4* dst = (float4*)atile;
#pragma unroll
    for (int j = 0; j < 8; ++j) {
      int fi = j * 32 + lane;              // float4 index; 16 per row
      float4 z; z.x = z.y = z.z = z.w = 0.f;
      if (r0 + (fi >> 4) < n_ent) z = src[fi];
      dst[fi] = z;
    }
  }
  __syncthreads();

  v8f acc = {};
  const v2f* bsw = (const v2f*)Bswz;
  const v2f* arow = (const v2f*)&atile[col * CDIM];
#pragma unroll
  for (int k4 = 0; k4 < 16; ++k4) {
    v2f a = arow[2 * k4 + khalf];          // A: K = 4*k4 + 2*khalf .. +1
    v2f b = bsw[k4 * 32 + lane];           // B: pre-swizzled
    acc = __builtin_amdgcn_wmma_f32_16x16x4_f32(false, a, false, b,
                                                (short)0, acc, false, false);
  }
#pragma unroll
  for (int v = 0; v < 8; ++v) {
    int r = r0 + v + 8 * khalf;
    if (r < n_ent) M[(size_t)r * 16 + col] = acc[v];
  }
}

// ---------------------------------------------------------------------------
// Segment-softmax helpers (scatter over `head`, n_ent segments)
// ---------------------------------------------------------------------------
__global__ void seg_init(int* __restrict__ mx1, int* __restrict__ mx2,
                         float* __restrict__ s1, float* __restrict__ s2,
                         float* __restrict__ cnt, int n) {
  long i = blockIdx.x * (long)blockDim.x + threadIdx.x;
  if (i < n) {
    mx1[i] = INT_MIN; mx2[i] = INT_MIN;
    s1[i] = 0.f; s2[i] = 0.f; cnt[i] = 0.f;
  }
}

__global__ void edge_score_pass1(const int* __restrict__ head,
                                 const int* __restrict__ etype,
                                 const float* __restrict__ M,
                                 float* __restrict__ score_r,
                                 int* __restrict__ segmax,
                                 float* __restrict__ cnt, int nE) {
  long e = blockIdx.x * (long)blockDim.x + threadIdx.x;
  if (e >= nE) return;
  int h = head[e];
  float s = M[(size_t)h * 16 + (etype[e] - 1)];
  score_r[e] = s;
  atomicMax(&segmax[h], f2ord(s));
  atomicAdd(&cnt[h], 1.0f);
}

__global__ void edge_exp_sum(const int* __restrict__ head,
                             const float* __restrict__ score,
                             const int* __restrict__ segmax,
                             float* __restrict__ ex,
                             float* __restrict__ segsum, int nE) {
  long e = blockIdx.x * (long)blockDim.x + threadIdx.x;
  if (e >= nE) return;
  int h = head[e];
  float v = expf(score[e] - ord2f(segmax[h]));
  ex[e] = v;
  atomicAdd(&segsum[h], v);
}

// wave per edge: score_trip = dot(ent[head],ent[tail]) + rel_score^2*||rel||^2
__global__ void edge_score_trip(const int* __restrict__ head,
                                const int* __restrict__ tail,
                                const int* __restrict__ etype,
                                const float* __restrict__ ent,
                                const float* __restrict__ ex,
                                const float* __restrict__ segsum,
                                const float* __restrict__ relnorm2,
                                float* __restrict__ strip,
                                int* __restrict__ segmax2, int nE) {
  long tid = blockIdx.x * (long)blockDim.x + threadIdx.x;
  long e = tid >> 5;
  int lane = threadIdx.x & 31;
  if (e >= nE) return;
  int eu = __builtin_amdgcn_readfirstlane((int)e);   // wave-uniform edge id
  int h = head[eu], t = tail[eu];
  float p = ent[(size_t)h * CDIM + lane] * ent[(size_t)t * CDIM + lane] +
            ent[(size_t)h * CDIM + lane + 32] * ent[(size_t)t * CDIM + lane + 32];
  for (int o = 16; o; o >>= 1) p += __shfl_xor(p, o, 32);
  if (lane == 0) {
    float rs = ex[eu] / (segsum[h] + 1e-16f);
    float st = p + rs * rs * relnorm2[etype[eu] - 1];
    strip[eu] = st;
    atomicMax(&segmax2[h], f2ord(st));
  }
}

__global__ void edge_mask(const int* __restrict__ head,
                          const float* __restrict__ ex2,
                          const float* __restrict__ segsum2,
                          float* __restrict__ kg_mask, int nE) {
  long e = blockIdx.x * (long)blockDim.x + threadIdx.x;
  if (e >= nE) return;
  kg_mask[e] = ex2[e] / (segsum2[head[e]] + 1e-16f);
}

// ---------------------------------------------------------------------------
// Hop kernels: one wave covers one edge (32 lanes x 2 channels, float2 path);
// edge index is wave-uniform -> scalarized via readfirstlane.
// ---------------------------------------------------------------------------
__global__ void kg_scatter(const float* __restrict__ ent,
                           const int* __restrict__ head,
                           const int* __restrict__ tail,
                           const int* __restrict__ etype,
                           const float* __restrict__ weight,
                           const float* __restrict__ kg_mask,
                           float* __restrict__ ent_agg, int nE) {
  long i = blockIdx.x * (long)blockDim.x + threadIdx.x;
  if (i >= (long)nE * 32) return;
  int e = __builtin_amdgcn_readfirstlane((int)(i >> 5));
  int c2 = (int)(i & 31) * 2;
  int h = head[e], t = tail[e], r = etype[e] - 1;
  float km = kg_mask[e];
  float2 tv = *(const float2*)&ent[(size_t)t * CDIM + c2];
  float2 wv = *(const float2*)&weight[r * CDIM + c2];
  atomicAdd(&ent_agg[(size_t)h * CDIM + c2],     tv.x * wv.x * km);
  atomicAdd(&ent_agg[(size_t)h * CDIM + c2 + 1], tv.y * wv.y * km);
}

__global__ void ui_scatter(const float* __restrict__ ent,
                           const int* __restrict__ rows,
                           const int* __restrict__ cols,
                           const float* __restrict__ vals,
                           float* __restrict__ u_agg, int nE) {
  long i = blockIdx.x * (long)blockDim.x + threadIdx.x;
  if (i >= (long)nE * 32) return;
  int e = __builtin_amdgcn_readfirstlane((int)(i >> 5));
  int c2 = (int)(i & 31) * 2;
  int rw = rows[e], cl = cols[e];
  float vl = vals[e];
  float2 cv = *(const float2*)&ent[(size_t)cl * CDIM + c2];
  atomicAdd(&u_agg[(size_t)rw * CDIM + c2],     cv.x * vl);
  atomicAdd(&u_agg[(size_t)rw * CDIM + c2 + 1], cv.y * vl);
}

__global__ void ent_update(const float* __restrict__ ent_agg,
                           const float* __restrict__ cnt,
                           float* __restrict__ ent_out,
                           float* __restrict__ ent_res, int n) {
  long tid = blockIdx.x * (long)blockDim.x + threadIdx.x;
  long i = tid >> 5;
  int lane = threadIdx.x & 31;
  if (i >= n) return;
  int iu = __builtin_amdgcn_readfirstlane((int)i);
  float cc = fmaxf(cnt[iu], 1.0f);
  float a = ent_agg[(size_t)iu * CDIM + lane] / cc;
  float b = ent_agg[(size_t)iu * CDIM + lane + 32] / cc;
  float ss = a * a + b * b;
  for (int o = 16; o; o >>= 1) ss += __shfl_xor(ss, o, 32);
  float inv = 1.0f / fmaxf(sqrtf(ss), 1e-12f);
  a *= inv; b *= inv;
  ent_out[(size_t)iu * CDIM + lane] = a;
  ent_out[(size_t)iu * CDIM + lane + 32] = b;
  ent_res[(size_t)iu * CDIM + lane] += a;
  ent_res[(size_t)iu * CDIM + lane + 32] += b;
}

__global__ void usr_update(const float* __restrict__ usr_in,
                           const float* __restrict__ u_agg,
                           const float* __restrict__ latent,
                           const float* __restrict__ disenw,
                           float* __restrict__ usr_out,
                           float* __restrict__ usr_res, int n) {
  long tid = blockIdx.x * (long)blockDim.x + threadIdx.x;
  long u = tid >> 5;
  int lane = threadIdx.x & 31;
  if (u >= n) return;
  int uu = __builtin_amdgcn_readfirstlane((int)u);
  float u0 = usr_in[(size_t)uu * CDIM + lane];
  float u1 = usr_in[(size_t)uu * CDIM + lane + 32];
  float d[NFACT];
  for (int f = 0; f < NFACT; ++f) {
    float p = u0 * latent[f * CDIM + lane] + u1 * latent[f * CDIM + lane + 32];
    for (int o = 16; o; o >>= 1) p += __shfl_xor(p, o, 32);
    d[f] = p;
  }
  float m = d[0];
  for (int f = 1; f < NFACT; ++f) m = fmaxf(m, d[f]);
  float s = 0.f;
  for (int f = 0; f < NFACT; ++f) { d[f] = expf(d[f] - m); s += d[f]; }
  float mix0 = 0.f, mix1 = 0.f;
  for (int f = 0; f < NFACT; ++f) {
    float w = d[f] / s;
    mix0 += w * disenw[f * CDIM + lane];
    mix1 += w * disenw[f * CDIM + lane + 32];
  }
  float a = u_agg[(size_t)uu * CDIM + lane] * (1.0f + mix0);
  float b = u_agg[(size_t)uu * CDIM + lane + 32] * (1.0f + mix1);
  float ss = a * a + b * b;
  for (int o = 16; o; o >>= 1) ss += __shfl_xor(ss, o, 32);
  float inv = 1.0f / fmaxf(sqrtf(ss), 1e-12f);
  a *= inv; b *= inv;
  usr_out[(size_t)uu * CDIM + lane] = a;
  usr_out[(size_t)uu * CDIM + lane + 32] = b;
  usr_res[(size_t)uu * CDIM + lane] += a;
  usr_res[(size_t)uu * CDIM + lane + 32] += b;
}

// ---------------------------------------------------------------------------
extern "C" void kernel_launch(void* const* d_in, const int* in_sizes, int n_in,
                              void* d_out, int out_size, void* d_ws, size_t ws_size,
                              hipStream_t stream) {
  (void)n_in; (void)out_size; (void)ws_size;
  const float* user_emb   = (const float*)d_in[0];
  const float* entity_emb = (const float*)d_in[1];
  const float* latent_emb = (const float*)d_in[2];
  const float* weight     = (const float*)d_in[3];
  const float* dwa        = (const float*)d_in[4];
  const float* kgW        = (const float*)d_in[5];
  const float* ui_vals    = (const float*)d_in[6];
  const int*   edge_index = (const int*)d_in[7];
  const int*   edge_type  = (const int*)d_in[8];
  const int*   ui_rows    = (const int*)d_in[9];
  const int*   ui_cols    = (const int*)d_in[10];

  const int n_usr = in_sizes[0] / CDIM;
  const int n_ent = in_sizes[1] / CDIM;
  const int nE    = in_sizes[8];
  const int* head = edge_index;
  const int* tail = edge_index + nE;

  float* out_ent = (float*)d_out;
  float* out_usr = out_ent + (size_t)n_ent * CDIM;
  float* out_cor = out_usr + (size_t)n_usr * CDIM;

  // workspace carve (256B aligned chunks)
  char* wptr = (char*)d_ws;
  auto alloc = [&](size_t bytes) -> char* {
    char* p = wptr;
    wptr += (bytes + 255) & ~(size_t)255;
    return p;
  };
  float* M        = (float*)alloc((size_t)n_ent * 16 * 4);
  float* Bswz     = (float*)alloc(16 * 32 * 2 * 4);
  float* relnorm2 = (float*)alloc(16 * 4);
  float* disenw   = (float*)alloc(NFACT * CDIM * 4);
  float* score_r  = (float*)alloc((size_t)nE * 4);
  float* exbuf    = (float*)alloc((size_t)nE * 4);
  float* strip    = (float*)alloc((size_t)nE * 4);
  float* kg_mask  = (float*)alloc((size_t)nE * 4);
  int*   segmax1  = (int*)alloc((size_t)n_ent * 4);
  int*   segmax2  = (int*)alloc((size_t)n_ent * 4);
  float* segsum1  = (float*)alloc((size_t)n_ent * 4);
  float* segsum2  = (float*)alloc((size_t)n_ent * 4);
  float* cnt      = (float*)alloc((size_t)n_ent * 4);
  float* ent_agg  = (float*)alloc((size_t)n_ent * CDIM * 4);
  float* u_agg    = (float*)alloc((size_t)n_usr * CDIM * 4);
  float* ent_cur  = (float*)alloc((size_t)n_ent * CDIM * 4);
  float* usr_cur  = (float*)alloc((size_t)n_usr * CDIM * 4);

  // residuals start at the input embeddings
  (void)hipMemcpyAsync(out_ent, entity_emb, (size_t)n_ent * CDIM * 4,
                       hipMemcpyDeviceToDevice, stream);
  (void)hipMemcpyAsync(out_usr, user_emb, (size_t)n_usr * CDIM * 4,
                       hipMemcpyDeviceToDevice, stream);

  precompute_small<<<1, 256, 0, stream>>>(weight, kgW, dwa, Bswz, relnorm2,
                                          disenw, out_cor);
  seg_init<<<(n_ent + 255) / 256, 256, 0, stream>>>(segmax1, segmax2, segsum1,
                                                    segsum2, cnt, n_ent);
  wmma_scores<<<(n_ent + 15) / 16, 32, 0, stream>>>(entity_emb, Bswz, M, n_ent);

  const int eb = (nE + 255) / 256;
  edge_score_pass1<<<eb, 256, 0, stream>>>(head, edge_type, M, score_r,
                                           segmax1, cnt, nE);
  edge_exp_sum<<<eb, 256, 0, stream>>>(head, score_r, segmax1, exbuf, segsum1, nE);
  const long wth = (long)nE * 32;
  edge_score_trip<<<(int)((wth + 255) / 256), 256, 0, stream>>>(
      head, tail, edge_type, entity_emb, exbuf, segsum1, relnorm2, strip,
      segmax2, nE);
  edge_exp_sum<<<eb, 256, 0, stream>>>(head, strip, segmax2, exbuf, segsum2, nE);
  edge_mask<<<eb, 256, 0, stream>>>(head, exbuf, segsum2, kg_mask, nE);

  const float* ent_in = entity_emb;
  const float* usr_in = user_emb;
  for (int hop = 0; hop < 2; ++hop) {
    (void)hipMemsetAsync(ent_agg, 0, (size_t)n_ent * CDIM * 4, stream);
    (void)hipMemsetAsync(u_agg, 0, (size_t)n_usr * CDIM * 4, stream);
    kg_scatter<<<(int)((wth + 255) / 256), 256, 0, stream>>>(
        ent_in, head, tail, edge_type, weight, kg_mask, ent_agg, nE);
    ui_scatter<<<(int)((wth + 255) / 256), 256, 0, stream>>>(
        ent_in, ui_rows, ui_cols, ui_vals, u_agg, nE);
    ent_update<<<(int)(((long)n_ent * 32 + 255) / 256), 256, 0, stream>>>(
        ent_agg, cnt, ent_cur, out_ent, n_ent);
    usr_update<<<(int)(((long)n_usr * 32 + 255) / 256), 256, 0, stream>>>(
        usr_in, u_agg, latent_emb, disenw, usr_cur, out_usr, n_usr);
    ent_in = ent_cur;
    usr_in = usr_cur;
  }
}